// ValueLayerBase_48266842472939
// MI455X (gfx1250) — compile-verified
//
#include <hip/hip_runtime.h>

// MI455X (gfx1250) — wave32, WMMA, NT streaming loads.
//
// Reference collapses algebraically: only wire-0's gates survive the Z-measurement
// (U = U0 (x) V, U^dag (Z(x)I) U = (U0^dag Z U0) (x) I), and the initial state is
// real, so per 1024-float sample:
//   z = [cos(rx0)cos(phi)*(S0-S1) - 2 sin(phi)*C] / (S0+S1),  phi = ry0[0]+ry1[0]
//   S0 = sum x[k]^2 (k<512), S1 = sum x[k]^2 (k>=512), C = sum x[k]*x[k+512]
// out[i, 0..15] = z_i  (rank-1 tile, produced with V_WMMA_F32_16X16X4_F32).
//
// Roofline: 64 MiB read + 1 MiB write @ 23.3 TB/s => ~2.9 us; ~100 MFLOP total.
// Purely bandwidth-bound => single coalesced NT streaming pass, one sample per wave32.

typedef __attribute__((ext_vector_type(2))) float v2f;
typedef __attribute__((ext_vector_type(4))) float v4f;
typedef __attribute__((ext_vector_type(8))) float v8f;

#define QDIM   1024
#define QHALF  512
#define SPB    16            // samples per block == waves per block
#define BLOCK  (SPB * 32)

__global__ __launch_bounds__(BLOCK)
void ValueLayer_quantum_z_kernel(const float* __restrict__ x,
                                 const float* __restrict__ rx0,
                                 const float* __restrict__ ry0,
                                 const float* __restrict__ ry1,
                                 float* __restrict__ out)
{
    __shared__ float zsh[SPB];

    const int lane = threadIdx.x & 31;
    const int wave = threadIdx.x >> 5;
    const long long sample = (long long)blockIdx.x * SPB + wave;
    const float* xs = x + sample * QDIM;

    // Stream one sample per wave: 8 x b128 NT loads, fully coalesced (512B/wave/load).
    v4f a[4], b[4];
#pragma unroll
    for (int i = 0; i < 4; ++i) {
        a[i] = __builtin_nontemporal_load((const v4f*)(xs + i * 128 + lane * 4));
        b[i] = __builtin_nontemporal_load((const v4f*)(xs + QHALF + i * 128 + lane * 4));
    }

    float s0 = 0.f, s1 = 0.f, cr = 0.f;
#pragma unroll
    for (int i = 0; i < 4; ++i) {
#pragma unroll
        for (int e = 0; e < 4; ++e) {
            const float av = a[i][e], bv = b[i][e];
            s0 = fmaf(av, av, s0);
            s1 = fmaf(bv, bv, s1);
            cr = fmaf(av, bv, cr);
        }
    }

    // wave32 butterfly reduction (warpSize == 32 on gfx1250)
#pragma unroll
    for (int m = 16; m; m >>= 1) {
        s0 += __shfl_xor(s0, m, 32);
        s1 += __shfl_xor(s1, m, 32);
        cr += __shfl_xor(cr, m, 32);
    }

    if (lane == 0) {
        const float thx = rx0[0];
        const float phi = ry0[0] + ry1[0];
        // |thx|,|phi| <~ 8 (sums of unit normals): hardware v_sin/v_cos is plenty
        // accurate (~5e-7 abs) and avoids the OCML Payne-Hanek slow path.
        const float kzz = __cosf(thx) * __cosf(phi);   // m00 = cos(theta)cos(phi)
        const float kxx = -2.0f * __sinf(phi);         // 2*Re(m01) = -2 sin(phi)
        zsh[wave] = (kzz * (s0 - s1) + kxx * cr) / (s0 + s1);
    }
    __syncthreads();

    // Rank-1 output tile via f32 WMMA: A[m][k] = z_m (all k), B[k][n] = 0.25
    // => D[m][n] = z_m * (4 * 0.25), exact in f32 and operand-half-layout agnostic.
    if (wave == 0) {
        const float zm = zsh[lane & 15];
        v2f A; A.x = zm;    A.y = zm;
        v2f B; B.x = 0.25f; B.y = 0.25f;
        v8f C = {};
        v8f D = __builtin_amdgcn_wmma_f32_16x16x4_f32(
            /*neg_a=*/false, A, /*neg_b=*/false, B,
            /*c_mod=*/(short)0, C, /*reuse_a=*/false, /*reuse_b=*/false);

        // C/D layout: VGPR j, lanes 0-15 -> (M=j, N=lane); lanes 16-31 -> (M=j+8, N=lane-16)
        const int n  = lane & 15;
        const int mb = (lane < 16) ? 0 : 8;
        float* o = out + (long long)blockIdx.x * (SPB * 16);
#pragma unroll
        for (int j = 0; j < 8; ++j)
            __builtin_nontemporal_store(D[j], o + (mb + j) * 16 + n);
    }
}

extern "C" void kernel_launch(void* const* d_in, const int* in_sizes, int n_in,
                              void* d_out, int out_size, void* d_ws, size_t ws_size,
                              hipStream_t stream) {
    const float* x   = (const float*)d_in[0];   // (8,2048,1024) f32
    const float* rx0 = (const float*)d_in[1];   // (10,) f32
    const float* ry0 = (const float*)d_in[2];   // (10,) f32
    const float* ry1 = (const float*)d_in[3];   // (10,) f32
    float* out = (float*)d_out;                 // (8,2048,16) f32

    const int N = in_sizes[0] / QDIM;           // 16384 samples
    const int blocks = N / SPB;                 // 1024 blocks of 512 threads (16 wave32)
    ValueLayer_quantum_z_kernel<<<blocks, BLOCK, 0, stream>>>(x, rx0, ry0, ry1, out);
}